// LocalSpatialEncoding_34239479284326
// MI455X (gfx1250) — compile-verified
//
#include <hip/hip_runtime.h>

typedef __attribute__((ext_vector_type(2))) float v2f;
typedef __attribute__((ext_vector_type(4))) float f4;
typedef __attribute__((ext_vector_type(8))) float v8f;

#define BN_EPS 1e-5f

// Each wave processes 32 unique samples per iteration (two 16-sample WMMA
// tiles). Per lane: load own neighbor + center, build enc=[rel,dist], run
// layer1 (4->4 + BN + ReLU) in VALU. A-fragments for the two
// v_wmma_f32_16x16x4_f32 tiles are assembled with ds_bpermute. Output staged
// through LDS and stored as two coalesced 512B b128 bursts per wave.
__global__ __launch_bounds__(256) void lse_wmma_kernel(
    const float* __restrict__ coords,   // (C,3) centers, C = B*N
    const float* __restrict__ nbr,      // (S,3) neighbors, S = B*N*K
    const float* __restrict__ W1,       // (4,4) row-major [o][c]
    const float* __restrict__ g1, const float* __restrict__ b1,
    const float* __restrict__ m1, const float* __restrict__ v1,
    const float* __restrict__ W2,       // (8,4) row-major [o][c]
    const float* __restrict__ g2, const float* __restrict__ b2,
    const float* __restrict__ m2, const float* __restrict__ v2,
    float* __restrict__ out,            // (S,8)
    int kShift, int nIters)
{
    __shared__ float smem[8 * 256];     // 8 waves * (32 samples * 8 ch)

    const int lane = threadIdx.x & 31;
    const int wib  = threadIdx.x >> 5;
    const int wid  = blockIdx.x * (blockDim.x >> 5) + wib;
    const int nWaves = gridDim.x * (blockDim.x >> 5);
    const int row = lane & 15;          // M row within a 16-sample tile
    const int hf  = lane >> 4;          // wave half -> K pair of A/B fragment

    // ---- loop-invariant parameters ----
    f4 w1r0 = *(const f4*)(W1 + 0);
    f4 w1r1 = *(const f4*)(W1 + 4);
    f4 w1r2 = *(const f4*)(W1 + 8);
    f4 w1r3 = *(const f4*)(W1 + 12);
    float s1[4], sh1[4];
#pragma unroll
    for (int o = 0; o < 4; ++o) {
        float sc = g1[o] * rsqrtf(v1[o] + BN_EPS);
        s1[o]  = sc;
        sh1[o] = b1[o] - m1[o] * sc;
    }
    // B fragment (shared by both tiles): B[k][n] = W2[n][k], col n = row,
    // cols >= 8 zero. Lanes 0-15: K=0(V0),1(V1); lanes 16-31: K=2(V0),3(V1).
    v2f bfrag;
    bfrag.x = (row < 8) ? W2[row * 4 + 2 * hf + 0] : 0.0f;
    bfrag.y = (row < 8) ? W2[row * 4 + 2 * hf + 1] : 0.0f;
    // BN2 params for output column n = row
    float s2 = 0.0f, sh2 = 0.0f;
    if (row < 8) {
        s2  = g2[row] * rsqrtf(v2[row] + BN_EPS);
        sh2 = b2[row] - m2[row] * s2;
    }

    float* sw = smem + wib * 256;
    const int rowB  = row * 4;          // bpermute byte index: lane `row`
    const int rowB2 = (row + 16) * 4;   // bpermute byte index: lane `row+16`

    for (int t = wid; t < nIters; t += nWaves) {
        const int base = t * 32;
        const int s    = base + lane;   // this lane's unique sample

        // prefetch next iteration's neighbor rows (speculative)
        __builtin_prefetch(nbr + (size_t)(s + nWaves * 32) * 3, 0, 1);

        // ---- encoding: rel coords + distance (own sample) ----
        const float* np = nbr + (size_t)s * 3;
        float nx = np[0], ny = np[1], nz = np[2];
        const float* cp = coords + (size_t)(s >> kShift) * 3;
        float dx = nx - cp[0];
        float dy = ny - cp[1];
        float dz = nz - cp[2];
        float dist = sqrtf(dx * dx + dy * dy + dz * dz);

        // ---- layer 1: 4x4 matmul + BN + ReLU (VALU) ----
        float h0 = dx * w1r0.x + dy * w1r0.y + dz * w1r0.z + dist * w1r0.w;
        float h1 = dx * w1r1.x + dy * w1r1.y + dz * w1r1.z + dist * w1r1.w;
        float h2 = dx * w1r2.x + dy * w1r2.y + dz * w1r2.z + dist * w1r2.w;
        float h3 = dx * w1r3.x + dy * w1r3.y + dz * w1r3.z + dist * w1r3.w;
        h0 = fmaxf(h0 * s1[0] + sh1[0], 0.0f);
        h1 = fmaxf(h1 * s1[1] + sh1[1], 0.0f);
        h2 = fmaxf(h2 * s1[2] + sh1[2], 0.0f);
        h3 = fmaxf(h3 * s1[3] + sh1[3], 0.0f);

        // ---- assemble A fragments for the two tiles via ds_bpermute ----
        // Tile0 (samples base+0..15): lane r -> {h0,h1} own;
        //                             lane 16+r -> {h2,h3} of lane r.
        float p2 = __int_as_float(__builtin_amdgcn_ds_bpermute(rowB, __float_as_int(h2)));
        float p3 = __int_as_float(__builtin_amdgcn_ds_bpermute(rowB, __float_as_int(h3)));
        v2f a0;
        a0.x = hf ? p2 : h0;
        a0.y = hf ? p3 : h1;
        // Tile1 (samples base+16..31): lane r -> {h0,h1} of lane 16+r;
        //                              lane 16+r -> {h2,h3} own.
        float p0 = __int_as_float(__builtin_amdgcn_ds_bpermute(rowB2, __float_as_int(h0)));
        float p1 = __int_as_float(__builtin_amdgcn_ds_bpermute(rowB2, __float_as_int(h1)));
        v2f a1;
        a1.x = hf ? h2 : p0;
        a1.y = hf ? h3 : p1;

        // ---- layer 2: two 16x4 @ 4x16 WMMAs ----
        v8f acc = {};
        v8f d0 = __builtin_amdgcn_wmma_f32_16x16x4_f32(
            false, a0, false, bfrag, (short)0, acc, false, false);
        v8f d1 = __builtin_amdgcn_wmma_f32_16x16x4_f32(
            false, a1, false, bfrag, (short)0, acc, false, false);

        // ---- BN2 + ReLU, stage both tiles into LDS in output layout ----
        // D element: VGPR g, lane -> (M = g + 8*hf, N = row)
        if (row < 8) {
#pragma unroll
            for (int g = 0; g < 8; ++g) {
                float v0 = fmaxf(d0[g] * s2 + sh2, 0.0f);
                float v1v = fmaxf(d1[g] * s2 + sh2, 0.0f);
                sw[(g + 8 * hf) * 8 + row]       = v0;
                sw[128 + (g + 8 * hf) * 8 + row] = v1v;
            }
        }
        __builtin_amdgcn_wave_barrier();   // LDS same-wave ops are in-order

        // ---- coalesced stores: 2 x (32 lanes x b128 = 512B contiguous) ----
        f4 ov0 = *(const f4*)(sw + lane * 4);
        f4 ov1 = *(const f4*)(sw + 128 + lane * 4);
        float* op = out + (size_t)base * 8;
        *(f4*)(op + lane * 4)       = ov0;
        *(f4*)(op + 128 + lane * 4) = ov1;
        __builtin_amdgcn_wave_barrier();   // protect LDS reuse next iteration
    }
}

extern "C" void kernel_launch(void* const* d_in, const int* in_sizes, int n_in,
                              void* d_out, int out_size, void* d_ws, size_t ws_size,
                              hipStream_t stream) {
    const float* coords = (const float*)d_in[0];
    const float* nbr    = (const float*)d_in[1];
    const float* W1     = (const float*)d_in[2];
    const float* g1     = (const float*)d_in[3];
    const float* b1     = (const float*)d_in[4];
    const float* m1     = (const float*)d_in[5];
    const float* v1     = (const float*)d_in[6];
    const float* W2     = (const float*)d_in[7];
    const float* g2     = (const float*)d_in[8];
    const float* b2     = (const float*)d_in[9];
    const float* m2     = (const float*)d_in[10];
    const float* v2     = (const float*)d_in[11];
    float* out = (float*)d_out;

    const int nSamples = in_sizes[1] / 3;      // B*N*K = 2,097,152
    const int nCenters = in_sizes[0] / 3;      // B*N
    const int Kn = nSamples / nCenters;        // 16 (power of two)
    int kShift = 0;
    while ((1 << kShift) < Kn) ++kShift;       // log2(K)

    const int nIters = nSamples / 32;          // 65536 32-sample groups

    const int threads = 256;                   // 8 wave32 per block
    int blocks = 1024;
    if (blocks * (threads / 32) > nIters) {
        blocks = (nIters + (threads / 32) - 1) / (threads / 32);
        if (blocks < 1) blocks = 1;
    }
    lse_wmma_kernel<<<blocks, threads, 0, stream>>>(
        coords, nbr, W1, g1, b1, m1, v1, W2, g2, b2, m2, v2,
        out, kShift, nIters);
}